// SolveFFT_14714557956189
// MI455X (gfx1250) — compile-verified
//
#include <hip/hip_runtime.h>

typedef __attribute__((ext_vector_type(2))) float v2f;
typedef __attribute__((ext_vector_type(8))) float v8f;

#define NB    2
#define COUT  64
#define CIN   3
#define HH    256
#define WF    129
#define NIMG  (NB*CIN)        // 6
#define XPAD  144             // Wf padded to multiple of 48 for GEMM-A N strips
#define KB    272             // 258 (Re|Im cols) padded to multiple of 16 for GEMM-B K dim
#define WOUT  256

// workspace layout (float offsets)
#define OFF_G   0
#define SZ_G    (512*512)           // stacked inverse-DFT matrix along H (row-major, lda=512)
#define OFF_B   (OFF_G + SZ_G)
#define SZ_B    (WOUT*KB)           // irfft-along-W matrix, TRANSPOSED: BT[x][k], ld=KB
#define OFF_XS  (OFF_B + SZ_B)
#define SZ_XS   (NIMG*XPAD*512)     // Xhat stack, TRANSPOSED: XT[img][n=w/pad][k=(h|256+h)], ld=512
#define OFF_T   (OFF_XS + SZ_XS)
#define SZ_T    (NIMG*WOUT*KB)      // intermediate T[img][h'][k], ld=KB

#define TWO_PI_OVER_256 0.024543692606170259758f

__global__ void sfft_zero_kernel(float* p, int n) {
    int i = blockIdx.x * blockDim.x + threadIdx.x;
    if (i < n) p[i] = 0.0f;
}

// G = [[C, -S], [S, C]],  C[h',h]=cos(2*pi*h*h'/256), S=sin(...)  (inverse DFT sign: +i)
__global__ void sfft_genG_kernel(float* G) {
    int idx = blockIdx.x * blockDim.x + threadIdx.x;
    if (idx >= 512 * 512) return;
    int r = idx >> 9;
    int c = idx & 511;
    int ph = ((r & 255) * (c & 255)) & 255;   // exact phase mod 256
    float th = (float)ph * TWO_PI_OVER_256;
    float cv = __cosf(th);
    float sv = __sinf(th);
    bool imrow = (r >= 256);
    bool imcol = (c >= 256);
    float v;
    if (!imrow)  v = imcol ? -sv : cv;
    else         v = imcol ?  cv : sv;
    G[idx] = v;
}

// Transposed irfft matrix: BT[x][k], k<129 -> wgt(w)*cos(2*pi*w*x/256)/65536 (w=k)
// 129<=k<258 -> -wgt(w)*sin(...)/65536 (w=k-129); k>=258 -> 0 (K padding, must be exact zero)
__global__ void sfft_genB_kernel(float* BT) {
    int idx = blockIdx.x * blockDim.x + threadIdx.x;
    if (idx >= WOUT * KB) return;
    int x = idx / KB;
    int k = idx % KB;
    float v = 0.0f;
    if (k < 258) {
        int w = (k < WF) ? k : (k - WF);
        float wgt = (w == 0 || w == 128) ? 1.0f : 2.0f;
        int p = (w * x) & 255;
        float th = (float)p * TWO_PI_OVER_256;
        float base = (k < WF) ? __cosf(th) : -__sinf(th);
        v = base * wgt * (1.0f / 65536.0f);
    }
    BT[idx] = v;
}

// One thread per (n,h,w) bin. Streams D (dominant, ~101MB) coalesced along w.
// Writes Xhat into the TRANSPOSED stack XT[img][w][h] (re) / XT[img][w][256+h] (im).
__global__ void sfft_solve_kernel(const float* __restrict__ X,
                                  const float* __restrict__ D,
                                  const float* __restrict__ Y,
                                  const float* __restrict__ alpha,
                                  float* __restrict__ XT) {
    int t = blockIdx.x * blockDim.x + threadIdx.x;
    if (t >= NB * HH * WF) return;
    int w = t % WF;
    int h = (t / WF) % HH;
    int n = t / (WF * HH);

    float a    = alpha[n] * (1.0f / (float)CIN);
    float inva = 1.0f / a;

    const float2* Xp = (const float2*)X;
    const float2* Dp = (const float2*)D;
    const float2* Yp = (const float2*)Y;

    float2 Xc[CIN];
#pragma unroll
    for (int c = 0; c < CIN; ++c)
        Xc[c] = Xp[((n * CIN + c) * HH + h) * WF + w];

    float accr[CIN] = {0.f, 0.f, 0.f};
    float acci[CIN] = {0.f, 0.f, 0.f};

    for (int co = 0; co < COUT; ++co) {
        float2 Yc = Yp[((n * COUT + co) * HH + h) * WF + w];
        float2 Dc[CIN], Zc[CIN];
        float numr = 0.f, numi = 0.f, den = 0.f;
#pragma unroll
        for (int c = 0; c < CIN; ++c) {
            Dc[c] = Dp[(((n * COUT + co) * CIN + c) * HH + h) * WF + w];
            float zr = Yc.x * Dc[c].x - Yc.y * Dc[c].y + a * Xc[c].x;
            float zi = Yc.x * Dc[c].y + Yc.y * Dc[c].x + a * Xc[c].y;
            Zc[c] = make_float2(zr, zi);
            numr += Dc[c].x * zr + Dc[c].y * zi;     // conj(D)*Z
            numi += Dc[c].x * zi - Dc[c].y * zr;
            den  += Dc[c].x * Dc[c].x + Dc[c].y * Dc[c].y;
        }
        float dinv = 1.0f / (a * den + a * a);       // denominator is purely real
        float sr = numr * dinv, si = numi * dinv;
#pragma unroll
        for (int c = 0; c < CIN; ++c) {
            accr[c] += Zc[c].x * inva - (Dc[c].x * sr - Dc[c].y * si);
            acci[c] += Zc[c].y * inva - (Dc[c].x * si + Dc[c].y * sr);
        }
    }

#pragma unroll
    for (int c = 0; c < CIN; ++c) {
        int img = n * CIN + c;
        float* base = XT + (size_t)img * XPAD * 512 + (size_t)w * 512;
        base[h]       = accr[c] * (1.0f / (float)COUT);
        base[256 + h] = acci[c] * (1.0f / (float)COUT);
    }
}

// GEMM-A: T = G(512x512) x Xhat(512x129) per image, WMMA f32 16x16x4.
// Each wave: one 16(M) x 48(N) strip -> 3 independent accumulators; A fragment reused 3x.
// B operand read K-contiguous from transposed XT (single b64 per lane per strip).
// Store remaps row r -> (h'=r&255, k = r<256 ? w : 129+w) into T[img][h'][k], w<129 only.
__global__ void sfft_gemmA_kernel(const float* __restrict__ G,
                                  const float* __restrict__ XT,
                                  float* __restrict__ T) {
    int m0  = blockIdx.x * 16;           // 0..511
    int n0  = blockIdx.y * 48;           // 0,48,96
    int img = blockIdx.z;

    int lane  = threadIdx.x;             // wave32, one wave per block
    int row   = lane & 15;
    int khalf = (lane >> 4) << 1;        // lanes 0-15: K=k0,k0+1 ; lanes 16-31: K=k0+2,k0+3

    const float* Bt = XT + (size_t)img * XPAD * 512;

    v8f acc0 = {0.f,0.f,0.f,0.f,0.f,0.f,0.f,0.f};
    v8f acc1 = acc0, acc2 = acc0;

    const float* bp0 = &Bt[(size_t)(n0 +  0 + row) * 512];
    const float* bp1 = &Bt[(size_t)(n0 + 16 + row) * 512];
    const float* bp2 = &Bt[(size_t)(n0 + 32 + row) * 512];
    const float* ap  = &G[(size_t)(m0 + row) * 512];

    for (int k0 = 0; k0 < 512; k0 += 4) {
        int kk = k0 + khalf;
        v2f aF = *(const v2f*)&ap[kk];
        v2f b0 = *(const v2f*)&bp0[kk];
        v2f b1 = *(const v2f*)&bp1[kk];
        v2f b2 = *(const v2f*)&bp2[kk];
        acc0 = __builtin_amdgcn_wmma_f32_16x16x4_f32(false, aF, false, b0, (short)0, acc0, false, false);
        acc1 = __builtin_amdgcn_wmma_f32_16x16x4_f32(false, aF, false, b1, (short)0, acc1, false, false);
        acc2 = __builtin_amdgcn_wmma_f32_16x16x4_f32(false, aF, false, b2, (short)0, acc2, false, false);
    }

    int mrow = m0 + ((lane >> 4) << 3);            // base M for this half-wave
    float* Tb = T + (size_t)img * WOUT * KB;
#pragma unroll
    for (int j = 0; j < 3; ++j) {
        int ncol = n0 + 16 * j + row;              // w
        if (ncol < WF) {
            v8f a = (j == 0) ? acc0 : (j == 1) ? acc1 : acc2;
#pragma unroll
            for (int v = 0; v < 8; ++v) {
                int r  = mrow + v;
                int hh = r & 255;
                int kk = (r < 256) ? ncol : (WF + ncol);
                Tb[hh * KB + kk] = a[v];
            }
        }
    }
}

// GEMM-B: Out(256x256) = T(256x272) x Birfft(272x256) per image -> d_out.
// Each wave: one 16(M) x 64(N) strip -> 4 independent accumulators; A fragment reused 4x.
// B operand read K-contiguous from transposed BT (single b64 per lane per strip).
__global__ void sfft_gemmB_kernel(const float* __restrict__ T,
                                  const float* __restrict__ BT,
                                  float* __restrict__ Out) {
    int m0  = blockIdx.x * 16;           // 0..255 (h')
    int n0  = blockIdx.y * 64;           // 0,64,128,192 (x)
    int img = blockIdx.z;

    int lane  = threadIdx.x;
    int row   = lane & 15;
    int khalf = (lane >> 4) << 1;

    const float* Ta = T + (size_t)img * WOUT * KB;
    float* Ob = Out + (size_t)img * WOUT * WOUT;

    v8f acc0 = {0.f,0.f,0.f,0.f,0.f,0.f,0.f,0.f};
    v8f acc1 = acc0, acc2 = acc0, acc3 = acc0;

    const float* ap  = &Ta[(size_t)(m0 + row) * KB];
    const float* bp0 = &BT[(size_t)(n0 +  0 + row) * KB];
    const float* bp1 = &BT[(size_t)(n0 + 16 + row) * KB];
    const float* bp2 = &BT[(size_t)(n0 + 32 + row) * KB];
    const float* bp3 = &BT[(size_t)(n0 + 48 + row) * KB];

    for (int k0 = 0; k0 < KB; k0 += 4) {
        int kk = k0 + khalf;
        v2f aF = *(const v2f*)&ap[kk];
        v2f b0 = *(const v2f*)&bp0[kk];
        v2f b1 = *(const v2f*)&bp1[kk];
        v2f b2 = *(const v2f*)&bp2[kk];
        v2f b3 = *(const v2f*)&bp3[kk];
        acc0 = __builtin_amdgcn_wmma_f32_16x16x4_f32(false, aF, false, b0, (short)0, acc0, false, false);
        acc1 = __builtin_amdgcn_wmma_f32_16x16x4_f32(false, aF, false, b1, (short)0, acc1, false, false);
        acc2 = __builtin_amdgcn_wmma_f32_16x16x4_f32(false, aF, false, b2, (short)0, acc2, false, false);
        acc3 = __builtin_amdgcn_wmma_f32_16x16x4_f32(false, aF, false, b3, (short)0, acc3, false, false);
    }

    int mrow = m0 + ((lane >> 4) << 3);
#pragma unroll
    for (int j = 0; j < 4; ++j) {
        v8f a = (j == 0) ? acc0 : (j == 1) ? acc1 : (j == 2) ? acc2 : acc3;
        int ncol = n0 + 16 * j + row;
#pragma unroll
        for (int v = 0; v < 8; ++v)
            Ob[(mrow + v) * WOUT + ncol] = a[v];
    }
}

extern "C" void kernel_launch(void* const* d_in, const int* in_sizes, int n_in,
                              void* d_out, int out_size, void* d_ws, size_t ws_size,
                              hipStream_t stream) {
    const float* X     = (const float*)d_in[0];
    const float* D     = (const float*)d_in[1];
    const float* Y     = (const float*)d_in[2];
    const float* alpha = (const float*)d_in[3];
    // d_in[4] = x_size (256,256) -- hardcoded
    (void)in_sizes; (void)n_in; (void)out_size; (void)ws_size;

    float* ws  = (float*)d_ws;
    float* out = (float*)d_out;

    // 1) zero XS (incl. N padding rows) + entire T (incl. K padding cols) -- contiguous
    {
        int n = SZ_XS + SZ_T;
        sfft_zero_kernel<<<(n + 255) / 256, 256, 0, stream>>>(ws + OFF_XS, n);
    }
    // 2) DFT matrices (generated each launch; deterministic)
    sfft_genG_kernel<<<(512 * 512) / 256, 256, 0, stream>>>(ws + OFF_G);
    sfft_genB_kernel<<<(WOUT * KB) / 256, 256, 0, stream>>>(ws + OFF_B);
    // 3) memory-bound Sherman-Morrison solve -> transposed [Re|Im] Xhat stack
    {
        int nbins = NB * HH * WF;               // 66048 = 258*256 exactly
        sfft_solve_kernel<<<(nbins + 255) / 256, 256, 0, stream>>>(X, D, Y, alpha, ws + OFF_XS);
    }
    // 4) inverse DFT along H (WMMA f32, 16x48 strip per wave)
    sfft_gemmA_kernel<<<dim3(512 / 16, XPAD / 48, NIMG), 32, 0, stream>>>(
        ws + OFF_G, ws + OFF_XS, ws + OFF_T);
    // 5) irfft along W (WMMA f32, 16x64 strip per wave) -> output
    sfft_gemmB_kernel<<<dim3(WOUT / 16, WOUT / 64, NIMG), 32, 0, stream>>>(
        ws + OFF_T, ws + OFF_B, out);
}